// AvgPool2dLayer_85074712199865
// MI455X (gfx1250) — compile-verified
//
#include <hip/hip_runtime.h>

// AvgPool 2x2 stride 2 over (B, 1024, 1024) f32 -> (B, 512, 512) f32.
// Bandwidth-bound (~0.2 flop/byte): data path is async global->LDS double
// buffering (CDNA5 GLOBAL_LOAD_ASYNC_TO_LDS_B128 + s_wait_asynccnt) with
// non-temporal b64 stores.

#define TPB 256
#define PAIRS_PER_WG 8
#define W_IN 1024
#define W_OUT 512

typedef float f32x4 __attribute__((ext_vector_type(4)));
typedef float f32x2 __attribute__((ext_vector_type(2)));
typedef int v4i __attribute__((ext_vector_type(4)));

// ---- CDNA5 async global->LDS copy (16B per lane) -------------------------
// Probe round 1 showed the builtin exists with signature
//   (int4* global_src, int4* lds_dst, imm offset, imm cpol)
// (generic-AS int4 pointers; void*/const failed to convert).
#if __has_builtin(__builtin_amdgcn_global_load_async_to_lds_b128)
#define ASYNC_CP16(g, l) \
  __builtin_amdgcn_global_load_async_to_lds_b128((v4i*)(g), (v4i*)(l), 0, 0)
#else
#define ASYNC_CP16(g, l)                                                      \
  do {                                                                        \
    unsigned __lds_off = (unsigned)(unsigned long long)(                      \
        __attribute__((address_space(3))) char*)(void*)(l);                   \
    asm volatile("global_load_async_to_lds_b128 %0, %1, off" ::"v"(__lds_off),\
                 "v"((const void*)(g))                                        \
                 : "memory");                                                 \
  } while (0)
#endif

#if __has_builtin(__builtin_amdgcn_s_wait_asynccnt)
#define WAIT_ASYNC(n) __builtin_amdgcn_s_wait_asynccnt(n)
#else
#define WAIT_ASYNC(n) asm volatile("s_wait_asynccnt " #n ::: "memory")
#endif

#define COMPILER_MEM_BARRIER() asm volatile("" ::: "memory")

// Main pipelined kernel: each WG handles PAIRS_PER_WG output rows (row-pairs
// of the input), full 1024-column width. Thread t owns input columns
// [4t, 4t+3] of both rows of a pair -> outputs [2t, 2t+1].
__global__ __launch_bounds__(TPB) void avgpool2x2_async(
    const float* __restrict__ in, float* __restrict__ out) {
  // [stage][row-in-pair][TPB * 4 floats] = 16 KB
  __shared__ __align__(16) float smem[2][2][TPB * 4];
  const int t = threadIdx.x;
  const long long base = (long long)blockIdx.x * PAIRS_PER_WG;

  // Prime the pipeline: pair `base` into stage 0.
  {
    const float* g = in + (base * 2) * W_IN + 4 * t;
    ASYNC_CP16(g, &smem[0][0][4 * t]);
    ASYNC_CP16(g + W_IN, &smem[0][1][4 * t]);
  }

#pragma unroll
  for (int i = 0; i < PAIRS_PER_WG; ++i) {
    const int s = i & 1;
    if (i + 1 < PAIRS_PER_WG) {
      // Prefetch next pair into the other stage, then retire current stage.
      const float* g = in + ((base + i + 1) * 2) * W_IN + 4 * t;
      ASYNC_CP16(g, &smem[s ^ 1][0][4 * t]);
      ASYNC_CP16(g + W_IN, &smem[s ^ 1][1][4 * t]);
      WAIT_ASYNC(2);  // async loads complete in order: stage s is ready
    } else {
      WAIT_ASYNC(0);
    }
    COMPILER_MEM_BARRIER();

    // Each thread reads back exactly the bytes its own async ops wrote:
    // no cross-thread LDS sharing, so no workgroup barrier is required.
    f32x4 a = *(const f32x4*)&smem[s][0][4 * t];
    f32x4 b = *(const f32x4*)&smem[s][1][4 * t];
    f32x2 o;
    o.x = (a.x + a.y + b.x + b.y) * 0.25f;
    o.y = (a.z + a.w + b.z + b.w) * 0.25f;
    __builtin_nontemporal_store(o, (f32x2*)(out + (base + i) * W_OUT) + t);
  }
}

// Tail kernel for leftover row-pairs (never launched for B=64, but keeps the
// launcher shape-robust). One pair per block, direct global loads.
__global__ __launch_bounds__(TPB) void avgpool2x2_tail(
    const float* __restrict__ in, float* __restrict__ out, long long p0) {
  const long long p = p0 + blockIdx.x;
  const int t = threadIdx.x;
  f32x4 a = *((const f32x4*)(in + (2 * p) * W_IN) + t);
  f32x4 b = *((const f32x4*)(in + (2 * p + 1) * W_IN) + t);
  f32x2 o;
  o.x = (a.x + a.y + b.x + b.y) * 0.25f;
  o.y = (a.z + a.w + b.z + b.w) * 0.25f;
  __builtin_nontemporal_store(o, (f32x2*)(out + p * W_OUT) + t);
}

extern "C" void kernel_launch(void* const* d_in, const int* in_sizes, int n_in,
                              void* d_out, int out_size, void* d_ws,
                              size_t ws_size, hipStream_t stream) {
  const float* x = (const float*)d_in[0];
  float* out = (float*)d_out;

  const long long n = in_sizes[0];          // B * 1024 * 1024
  const long long B = n >> 20;              // H = W = 1024 fixed by reference
  const long long total_pairs = B * W_OUT;  // one pair per output row

  const long long full_blocks = total_pairs / PAIRS_PER_WG;
  if (full_blocks > 0) {
    avgpool2x2_async<<<(unsigned)full_blocks, TPB, 0, stream>>>(x, out);
  }
  const long long rem = total_pairs - full_blocks * PAIRS_PER_WG;
  if (rem > 0) {
    avgpool2x2_tail<<<(unsigned)rem, TPB, 0, stream>>>(
        x, out, full_blocks * PAIRS_PER_WG);
  }
}